// PitchRegisterTracker_21741124452495
// MI455X (gfx1250) — compile-verified
//
#include <hip/hip_runtime.h>
#include <stdint.h>

#define CHUNK 32768   // elements per block in count/tail passes
#define TILE  4096    // floats per TDM tile (16 KB LDS)

typedef unsigned int u32;
typedef u32 u32x4 __attribute__((ext_vector_type(4)));
typedef int i32x4 __attribute__((ext_vector_type(4)));
typedef int i32x8 __attribute__((ext_vector_type(8)));
typedef float f32x4 __attribute__((ext_vector_type(4)));   // clang ext_vector: valid for NT store

#if defined(__AMDGCN__) && __has_builtin(__builtin_amdgcn_tensor_load_to_lds) && __has_builtin(__builtin_amdgcn_s_wait_tensorcnt)
#define HAVE_TDM 1
#else
#define HAVE_TDM 0
#endif

#if HAVE_TDM
// Issue a TDM DMA of `len` floats from `src` into LDS at `lds_addr` (zero-fill past len).
// Must be called by a full wave (TDM ignores EXEC; we branch whole waves around it).
__device__ __forceinline__ void tdm_load_tile(const float* src, int len, u32 lds_addr) {
  u32x4 g0;
  g0[0] = 1u;                                        // count=1, user mode, no gather
  g0[1] = lds_addr;                                  // LDS byte address
  uint64_t ga = (uint64_t)(size_t)src;               // global tile start
  g0[2] = (u32)ga;
  g0[3] = ((u32)(ga >> 32) & 0x01FFFFFFu) | (2u << 30);  // addr[56:32] | type=2
  const u32 td0 = (u32)len;                          // tensor_dim0 = len (OOB -> zero)
  i32x8 g1;
  g1[0] = (int)(2u << 16);                           // data_size=2 (4 bytes)
  g1[1] = (int)((td0 & 0xFFFFu) << 16);              // tensor_dim0[15:0]
  g1[2] = (int)(((td0 >> 16) & 0xFFFFu) | (1u << 16)); // tensor_dim0[31:16], tensor_dim1=1
  g1[3] = (int)(((u32)TILE) << 16);                  // tile_dim0 = TILE
  g1[4] = 1;                                         // tile_dim1 = 1, tile_dim2 = 0
  g1[5] = (int)(u32)TILE;                            // tensor_dim0_stride lo
  g1[6] = 0; g1[7] = 0;
  i32x4 g2 = {0, 0, 0, 0};
  i32x4 g3 = {0, 0, 0, 0};
#if __clang_major__ >= 23
  i32x8 g4 = {0, 0, 0, 0, 0, 0, 0, 0};
  __builtin_amdgcn_tensor_load_to_lds(g0, g1, g2, g3, g4, 0);
#else
  __builtin_amdgcn_tensor_load_to_lds(g0, g1, g2, g3, 0);
#endif
}
#endif

// ---------------- Pass 4: finalize (pipelined TDM stream -> LDS -> NT b128 stores) ---
// TDM for the first tile is issued BEFORE the stats prologue so the DMA overlaps the
// fp32 divide/sqrt sequence; the loop re-issues into the same buffer after the
// post-processing barrier.
__global__ void __launch_bounds__(256) k_final(const float* __restrict__ in,
                                               float* __restrict__ out, long n,
                                               const float* __restrict__ sums,
                                               const int* __restrict__ meta) {
  __shared__ float tile[TILE];
  const long nTiles = (n + TILE - 1) / TILE;
  long tb = blockIdx.x;

#if HAVE_TDM
  if (tb < nTiles && threadIdx.x < 32) {             // wave 0: kick off first DMA now
    const long base0 = tb * (long)TILE;
    long rem0 = n - base0;
    const int len0 = (int)((rem0 < (long)TILE) ? rem0 : (long)TILE);
    tdm_load_tile(in + base0, len0, (u32)(size_t)(&tile[0]));
  }
#endif

  // stats prologue (overlaps the in-flight DMA)
  const int count = meta[2];
  const float s0 = sums[0], s1 = sums[1];
  float mean = (count > 0) ? (s0 / (float)count) : 0.f;
  int dn = count - 1; if (dn < 1) dn = 1;
  float var = (s1 - (float)count * mean * mean) / (float)dn;
  float sd = (count > 1) ? sqrtf(fmaxf(var, 0.f)) : 1.f;
  sd = fmaxf(sd, 1e-7f);
  const float TLS = 40.0f / (200.0f * 0.693147f);    // TARGET_LOG_STD
  const float A = TLS / sd;
  const float B = 7.6438561897747395f - mean * A;    // log2(200) - mean*A

  while (tb < nTiles) {
    const long base = tb * (long)TILE;
    long rem = n - base;
    const int len = (int)((rem < (long)TILE) ? rem : (long)TILE);
#if HAVE_TDM
    if (threadIdx.x < 32) __builtin_amdgcn_s_wait_tensorcnt(0);
    __syncthreads();
#else
    for (int i = threadIdx.x; i < TILE; i += blockDim.x) {
      long j = base + i;
      tile[i] = (j < n) ? in[j] : 0.f;
    }
    __syncthreads();
#endif
    // vectorized elementwise map from LDS -> global; NT stores keep input hot in L2
    for (int i = threadIdx.x * 4; i + 3 < len; i += blockDim.x * 4) {
      f32x4 v = *reinterpret_cast<const f32x4*>(&tile[i]);
      f32x4 r;
      r.x = (v.x > 0.f) ? exp2f(__log2f(v.x) * A + B) : 0.f;
      r.y = (v.y > 0.f) ? exp2f(__log2f(v.y) * A + B) : 0.f;
      r.z = (v.z > 0.f) ? exp2f(__log2f(v.z) * A + B) : 0.f;
      r.w = (v.w > 0.f) ? exp2f(__log2f(v.w) * A + B) : 0.f;
      __builtin_nontemporal_store(r, reinterpret_cast<f32x4*>(out + base + i));
    }
    const int lenr = len & ~3;
    for (int i = lenr + threadIdx.x; i < len; i += blockDim.x) {
      float v = tile[i];
      float r = (v > 0.f) ? exp2f(__log2f(v) * A + B) : 0.f;
      __builtin_nontemporal_store(r, out + base + i);
    }
    const long nxt = tb + gridDim.x;
    __syncthreads();                                 // everyone done reading tile
#if HAVE_TDM
    if (nxt < nTiles && threadIdx.x < 32) {          // wave 0: next DMA into same buffer
      const long baseN = nxt * (long)TILE;
      long remN = n - baseN;
      const int lenN = (int)((remN < (long)TILE) ? remN : (long)TILE);
      tdm_load_tile(in + baseN, lenN, (u32)(size_t)(&tile[0]));
    }
#endif
    tb = nxt;
  }
}

// ---------------- Pass 1: per-block valid counts (streams input once) ----------------
__global__ void __launch_bounds__(256) k_count(const float* __restrict__ in, long n,
                                               int* __restrict__ cnts) {
  __shared__ int wsum[8];
  const int t = threadIdx.x;
  const long base = (long)blockIdx.x * CHUNK;
  int local = 0;
  if (base + CHUNK <= n) {
    // fast path: whole chunk in range, unconditional float4 streaming
    const f32x4* p = reinterpret_cast<const f32x4*>(in + base) + t;
#pragma unroll 8
    for (int it = 0; it < CHUNK / (256 * 4); ++it) {   // 32 iters of float4
      const f32x4 v = p[it * 256];
      local += (v.x > 0.f) + (v.y > 0.f) + (v.z > 0.f) + (v.w > 0.f);
    }
  } else {
    for (int it = 0; it < CHUNK / 256; ++it) {
      long j = base + it * 256 + t;
      if (j < n) local += (in[j] > 0.f);
    }
  }
  for (int off = 16; off > 0; off >>= 1) local += __shfl_down(local, off, 32);
  if ((t & 31) == 0) wsum[t >> 5] = local;
  __syncthreads();
  if (t == 0) {
    int s = 0;
    for (int w = 0; w < 8; ++w) s += wsum[w];
    cnts[blockIdx.x] = s;
  }
}

// ---------------- Pass 2: exclusive scan of block counts + meta (1 workgroup) --------
__global__ void __launch_bounds__(1024) k_scan(int* __restrict__ cnts, int nb,
                                               int* __restrict__ meta,
                                               float* __restrict__ sums) {
  __shared__ int s[1024];
  const int t = threadIdx.x;
  int P = (nb + 1023) >> 10;
  if (P > 16) P = 16;                      // supports nb <= 16384 (N <= 512M)
  int loc[16];
  int tot = 0;
  for (int i = 0; i < P; ++i) {
    int idx = t * P + i;
    int c = (idx < nb) ? cnts[idx] : 0;
    loc[i] = tot; tot += c;
  }
  s[t] = tot;
  __syncthreads();
  for (int off = 1; off < 1024; off <<= 1) {   // Hillis-Steele inclusive scan
    int v = (t >= off) ? s[t - off] : 0;
    __syncthreads();
    s[t] += v;
    __syncthreads();
  }
  const int excl = (t == 0) ? 0 : s[t - 1];
  for (int i = 0; i < P; ++i) {
    int idx = t * P + i;
    if (idx < nb) cnts[idx] = excl + loc[i];   // overwrite with exclusive prefix
  }
  if (t == 1023) {
    int nvalid = s[1023];
    cnts[nb] = nvalid;
    int count = nvalid < 1000 ? nvalid : 1000;
    meta[0] = nvalid;
    meta[1] = nvalid - count;                  // R0: first rank inside the window
    meta[2] = count;
    sums[0] = 0.f;                             // zero accumulators every call
    sums[1] = 0.f;
  }
}

// ---------------- Pass 3: accumulate sum / sumsq over ranks >= R0 --------------------
__global__ void __launch_bounds__(256) k_tail(const float* __restrict__ in, long n,
                                              const int* __restrict__ prefix,
                                              const int* __restrict__ meta,
                                              float* __restrict__ sums) {
  const int b = blockIdx.x;
  const int p0 = prefix[b], p1 = prefix[b + 1];
  const int R0 = meta[1];
  if (p1 <= R0) return;                        // no window ranks in this chunk
  __shared__ int wcnt[8];
  __shared__ float red[16];
  const int t = threadIdx.x, wave = t >> 5, lane = t & 31;
  const long base = (long)b * CHUNK;
  int running = p0;
  float lsum = 0.f, lsq = 0.f;
  for (int it = 0; it < CHUNK / 256; ++it) {   // order-preserving rank scan
    long e = base + it * 256 + t;
    float v = (e < n) ? in[e] : 0.f;
    bool pred = v > 0.f;
    unsigned mask = (unsigned)__ballot(pred);  // wave32: low 32 bits
    if (lane == 0) wcnt[wave] = __popc(mask);
    __syncthreads();
    int woff = 0, ttot = 0;
    for (int w = 0; w < 8; ++w) { int c = wcnt[w]; if (w < wave) woff += c; ttot += c; }
    if (pred) {
      int rank = running + woff + __popc(mask & ((1u << lane) - 1u));
      if (rank >= R0) { float lp = __log2f(v); lsum += lp; lsq += lp * lp; }
    }
    running += ttot;
    __syncthreads();
  }
  for (int off = 16; off > 0; off >>= 1) {
    lsum += __shfl_down(lsum, off, 32);
    lsq  += __shfl_down(lsq, off, 32);
  }
  if (lane == 0) { red[wave] = lsum; red[8 + wave] = lsq; }
  __syncthreads();
  if (t == 0) {
    float a = 0.f, q = 0.f;
    for (int w = 0; w < 8; ++w) { a += red[w]; q += red[8 + w]; }
    atomicAdd(&sums[0], a);
    atomicAdd(&sums[1], q);
  }
}

extern "C" void kernel_launch(void* const* d_in, const int* in_sizes, int n_in,
                              void* d_out, int out_size, void* d_ws, size_t ws_size,
                              hipStream_t stream) {
  const float* in = (const float*)d_in[0];
  float* out = (float*)d_out;
  const long n = (long)in_sizes[0];
  const int nb = (int)((n + CHUNK - 1) / CHUNK);       // 1024 blocks for N=2^25

  int* ws_i = (int*)d_ws;
  int* meta = ws_i;                  // [0]=n_valid [1]=R0 [2]=count
  float* sums = (float*)(ws_i + 4);  // [0]=sum(log2) [1]=sum(log2^2)
  int* prefix = ws_i + 8;            // nb+1 ints (counts, then exclusive prefix)

  k_count<<<nb, 256, 0, stream>>>(in, n, prefix);
  k_scan<<<1, 1024, 0, stream>>>(prefix, nb, meta, sums);
  k_tail<<<nb, 256, 0, stream>>>(in, n, prefix, meta, sums);

  long nTiles = (n + TILE - 1) / TILE;
  int gridD = (int)((nTiles < 16384) ? nTiles : 16384);
  if (gridD < 1) gridD = 1;
  k_final<<<gridD, 256, 0, stream>>>(in, out, n, sums, meta);
}